// FraudGNN_12575664242835
// MI455X (gfx1250) — compile-verified
//
#include <hip/hip_runtime.h>

typedef __attribute__((ext_vector_type(2))) float v2f;
typedef __attribute__((ext_vector_type(8))) float v8f;

#define DIM 64   // feature dim for both layers

// ---------------------------------------------------------------- zero
__global__ void zero_f32(float* __restrict__ p, long long n) {
  long long i = (long long)blockIdx.x * blockDim.x + threadIdx.x;
  long long stride = (long long)gridDim.x * blockDim.x;
  for (; i < n; i += stride) p[i] = 0.0f;
}

// ---------------------------------------------------------------- degree
__global__ void degree_kernel(const int* __restrict__ dst,
                              float* __restrict__ deg, int E) {
  int e = blockIdx.x * blockDim.x + threadIdx.x;
  if (e < E) atomicAdd(&deg[dst[e]], 1.0f);
}

// ------------------------------------------------- edge scatter (mean-sum)
// One wave32 per edge; each lane moves 2 of the 64 features.
__global__ void scatter_kernel(const float* __restrict__ feat,
                               const int* __restrict__ src,
                               const int* __restrict__ dst,
                               float* __restrict__ agg, int E) {
  long long t = (long long)blockIdx.x * blockDim.x + threadIdx.x;
  int e = (int)(t >> 5);
  int lane = (int)(t & 31);
  if (e >= E) return;
  int s = src[e];
  int d = dst[e];
  const float2 v = *(const float2*)(feat + (size_t)s * DIM + lane * 2);
  float* o = agg + (size_t)d * DIM + lane * 2;
  atomicAdd(o, v.x);
  atomicAdd(o + 1, v.y);
}

// ------------------------------------------------- fused SAGE linear (WMMA)
// out[i,:] = act( (agg[i,:]/max(deg,1)) @ Wl^T + bl + xin[i,:] @ Wr^T )
// One wave computes a 16-node x 64-output tile via V_WMMA_F32_16X16X4_F32,
// K = 128 (concat of mean-part and self-part). Weights staged in LDS.
template <bool RELU>
__global__ void __launch_bounds__(256)
sage_linear_kernel(const float* __restrict__ agg,
                   const float* __restrict__ deg,
                   const float* __restrict__ xin,
                   const float* __restrict__ Wl,
                   const float* __restrict__ bl,
                   const float* __restrict__ Wr,
                   float* __restrict__ out, int N) {
  __shared__ float ws[DIM * 2 * DIM]; // [64 out-rows][128 concat-cols]
  __shared__ float bs[DIM];

  // Stage concatenated weights: ws[j][k] = (k<64) ? Wl[j][k] : Wr[j][k-64]
  for (int f = threadIdx.x; f < DIM * 2 * DIM; f += blockDim.x) {
    int r = f >> 7;       // output feature j
    int c = f & 127;      // concat input k
    ws[f] = (c < DIM) ? Wl[r * DIM + c] : Wr[r * DIM + (c - DIM)];
  }
  if (threadIdx.x < DIM) bs[threadIdx.x] = bl[threadIdx.x];
  __syncthreads();

  const int wave = threadIdx.x >> 5;
  const int lane = threadIdx.x & 31;
  const int col  = lane & 15;  // A-row index / B-col index / D-col index
  const int hi   = lane >> 4;  // K sub-select for A/B, M>=8 select for D
  const int node_base = (blockIdx.x * 8 + wave) * 16;

  int nrow = node_base + col;  // node feeding this lane's A rows
  if (nrow > N - 1) nrow = N - 1;   // clamp tail (EXEC stays all-ones)

  const float inv = 1.0f / fmaxf(deg[nrow], 1.0f);
  const float* arow = agg + (size_t)nrow * DIM;
  const float* xrow = xin + (size_t)nrow * DIM;

  v8f acc[4];
  acc[0] = (v8f){}; acc[1] = (v8f){}; acc[2] = (v8f){}; acc[3] = (v8f){};

  // K = 0..63 : mean-aggregated part (scaled by 1/deg)
  #pragma unroll
  for (int kk = 0; kk < DIM; kk += 4) {
    const int kb = kk + hi * 2;
    float2 a2 = *(const float2*)(arow + kb);
    v2f av; av[0] = a2.x * inv; av[1] = a2.y * inv;
    #pragma unroll
    for (int c = 0; c < 4; ++c) {
      v2f bv = *(const v2f*)(ws + (size_t)(c * 16 + col) * 128 + kb);
      acc[c] = __builtin_amdgcn_wmma_f32_16x16x4_f32(
          false, av, false, bv, (short)0, acc[c], false, false);
    }
  }
  // K = 64..127 : self (root) part
  #pragma unroll
  for (int kk = DIM; kk < 2 * DIM; kk += 4) {
    const int kb = kk + hi * 2;
    float2 a2 = *(const float2*)(xrow + (kb - DIM));
    v2f av; av[0] = a2.x; av[1] = a2.y;
    #pragma unroll
    for (int c = 0; c < 4; ++c) {
      v2f bv = *(const v2f*)(ws + (size_t)(c * 16 + col) * 128 + kb);
      acc[c] = __builtin_amdgcn_wmma_f32_16x16x4_f32(
          false, av, false, bv, (short)0, acc[c], false, false);
    }
  }

  // D layout: VGPR R holds M=R (lanes 0-15) and M=R+8 (lanes 16-31), N=col
  #pragma unroll
  for (int c = 0; c < 4; ++c) {
    const float bias = bs[c * 16 + col];
    #pragma unroll
    for (int R = 0; R < 8; ++R) {
      int orow = node_base + R + 8 * hi;
      if (orow < N) {
        float v = acc[c][R] + bias;
        if (RELU) v = fmaxf(v, 0.0f);
        out[(size_t)orow * DIM + c * 16 + col] = v;
      }
    }
  }
}

// ---------------------------------------------------------------- classifier
__global__ void classifier_kernel(const float* __restrict__ h,
                                  const float* __restrict__ Wc,
                                  const float* __restrict__ bc,
                                  float* __restrict__ out, int N) {
  int i = blockIdx.x * blockDim.x + threadIdx.x;
  if (i >= N) return;
  const float* row = h + (size_t)i * DIM;
  float a0 = 0.0f, a1 = 0.0f;
  #pragma unroll
  for (int k = 0; k < DIM; k += 4) {
    float4 v  = *(const float4*)(row + k);
    float4 w0 = *(const float4*)(Wc + k);
    float4 w1 = *(const float4*)(Wc + DIM + k);
    a0 += v.x * w0.x + v.y * w0.y + v.z * w0.z + v.w * w0.w;
    a1 += v.x * w1.x + v.y * w1.y + v.z * w1.z + v.w * w1.w;
  }
  out[(size_t)i * 2 + 0] = a0 + bc[0];
  out[(size_t)i * 2 + 1] = a1 + bc[1];
}

// ---------------------------------------------------------------- launch
extern "C" void kernel_launch(void* const* d_in, const int* in_sizes, int n_in,
                              void* d_out, int out_size, void* d_ws, size_t ws_size,
                              hipStream_t stream) {
  const float* x   = (const float*)d_in[0];
  const int*   ei  = (const int*)d_in[1];
  const float* Wl1 = (const float*)d_in[2];
  const float* bl1 = (const float*)d_in[3];
  const float* Wr1 = (const float*)d_in[4];
  const float* Wl2 = (const float*)d_in[5];
  const float* bl2 = (const float*)d_in[6];
  const float* Wr2 = (const float*)d_in[7];
  const float* Wc  = (const float*)d_in[8];
  const float* bc  = (const float*)d_in[9];

  const int N = in_sizes[0] / DIM;   // 100000
  const int E = in_sizes[1] / 2;     // 1200000
  const int* src = ei;
  const int* dst = ei + E;

  // Workspace layout (floats): deg[N] | agg[N*64] | h1[N*64] | h2[N*64]
  float* deg = (float*)d_ws;
  float* aggb = deg + N;
  float* h1  = aggb + (size_t)N * DIM;
  float* h2  = h1 + (size_t)N * DIM;

  const int ZB = 2048;
  // zero deg + agg in one pass (contiguous)
  zero_f32<<<ZB, 256, 0, stream>>>(deg, (long long)N * (DIM + 1));

  degree_kernel<<<(E + 255) / 256, 256, 0, stream>>>(dst, deg, E);

  const long long sthreads = (long long)E * 32;
  const int sBlocks = (int)((sthreads + 255) / 256);
  scatter_kernel<<<sBlocks, 256, 0, stream>>>(x, src, dst, aggb, E);

  const int linBlocks = (N + 127) / 128;  // 128 nodes per 256-thread block
  sage_linear_kernel<true><<<linBlocks, 256, 0, stream>>>(
      aggb, deg, x, Wl1, bl1, Wr1, h1, N);

  zero_f32<<<ZB, 256, 0, stream>>>(aggb, (long long)N * DIM);
  scatter_kernel<<<sBlocks, 256, 0, stream>>>(h1, src, dst, aggb, E);
  sage_linear_kernel<false><<<linBlocks, 256, 0, stream>>>(
      aggb, deg, h1, Wl2, bl2, Wr2, h2, N);

  classifier_kernel<<<(N + 255) / 256, 256, 0, stream>>>(h2, Wc, bc,
                                                         (float*)d_out, N);
}